// hd_classifier_11115375362816
// MI455X (gfx1250) — compile-verified
//
#include <hip/hip_runtime.h>
#include <float.h>

// Problem constants (match reference)
#define NCLASSES 26
#define DHV      10000   // D
#define NROWS    8192    // N
// ALPHA == 1.0 -> new = chv + segsum(targets) - segsum(preds)

typedef __attribute__((ext_vector_type(2))) float v2f;
typedef __attribute__((ext_vector_type(8))) float v8f;

#define DPB       256   // dims per block in update pass
#define ROWCHUNKS 16    // row-split for update pass

// ---------------------------------------------------------------------------
// Kernel 0: zero the [NCLASSES, D] f32 accumulator in workspace
// ---------------------------------------------------------------------------
__global__ void zero_acc_kernel(float* __restrict__ acc, int n) {
    int i = blockIdx.x * blockDim.x + threadIdx.x;
    if (i < n) acc[i] = 0.0f;
}

// ---------------------------------------------------------------------------
// Kernel 1: scores = enc @ chv^T via V_WMMA_F32_16X16X4_F32, fused per-row
// argmax -> preds.  One wave computes a 16-row x 32-class tile.  Classes
// 26..31 are *not* zeroed: their B pointer is clamped to class row 0 and the
// resulting garbage accumulator columns are simply never read (score store is
// guarded, argmax substitutes -FLT_MAX).  Inner loop is pure loads + WMMA.
// Unroll factor 10 divides the trip count (2500) exactly -> no remainder
// peeling, no accumulator register shuffling.
//
// A frag (16x4 f32, 2 VGPR): lane = m + 16*(k>=2), vgpr = k&1
//   -> lane loads float2 enc[row=m][kt*4 + 2*kh + {0,1}]
// B frag (4x16 f32, 2 VGPR): lane = n + 16*(k>=2), vgpr = k&1
//   -> lane loads float2 chv[class=n][kt*4 + 2*kh + {0,1}]
// C/D (16x16 f32, 8 VGPR): vgpr r -> row r + 8*kh, col = m
// ---------------------------------------------------------------------------
__global__ void __launch_bounds__(256)
gemm_scores_preds_kernel(const float* __restrict__ enc,
                         const float* __restrict__ chv,
                         float* __restrict__ scores,
                         int*   __restrict__ preds) {
    const int wave    = threadIdx.x >> 5;
    const int lane    = threadIdx.x & 31;
    const int rowtile = blockIdx.x * (blockDim.x >> 5) + wave;   // 0..511
    const int m  = lane & 15;     // A row within tile / B class column
    const int kh = lane >> 4;     // K-half select

    const float* aptr  = enc + (size_t)(rowtile * 16 + m) * DHV + 2 * kh;
    const float* b0ptr = chv + (size_t)m * DHV + 2 * kh;
    const int   n1     = 16 + m;
    const bool  v1     = (n1 < NCLASSES);
    // clamp invalid class rows to row 0; those accumulator columns are dead
    const float* b1ptr = chv + (size_t)(v1 ? n1 : 0) * DHV + 2 * kh;

    v8f acc0 = {};   // classes 0..15
    v8f acc1 = {};   // classes 16..25 (+6 dead pad columns)

#pragma unroll 10
    for (int kt = 0; kt < DHV / 4; ++kt) {
        v2f a  = *(const v2f*)(aptr  + kt * 4);
        v2f b0 = *(const v2f*)(b0ptr + kt * 4);
        v2f b1 = *(const v2f*)(b1ptr + kt * 4);
        // hint the A stream ahead (lowers to global_prefetch_b8)
        __builtin_prefetch(aptr + kt * 4 + 1024, 0, 0);
        acc0 = __builtin_amdgcn_wmma_f32_16x16x4_f32(
            false, a, false, b0, (short)0, acc0, false, false);
        acc1 = __builtin_amdgcn_wmma_f32_16x16x4_f32(
            false, a, false, b1, (short)0, acc1, false, false);
    }

    // Write scores + fused argmax.  Each 16-lane half holds all classes for
    // rows (r + 8*kh), r = 0..7: lane's column is class m (acc0) / 16+m (acc1).
    const int rowbase = rowtile * 16 + 8 * kh;
#pragma unroll
    for (int r = 0; r < 8; ++r) {
        const int row = rowbase + r;
        scores[(size_t)row * NCLASSES + m] = acc0[r];
        if (v1) scores[(size_t)row * NCLASSES + n1] = acc1[r];

        // lane-local best (prefer lower class index on ties, like jnp.argmax)
        float bv = acc0[r];
        int   bi = m;
        float av = v1 ? acc1[r] : -FLT_MAX;   // dead pad columns masked here
        if (av > bv) { bv = av; bi = n1; }
        // butterfly across the 16-lane half (wave32, offsets stay in-half)
#pragma unroll
        for (int off = 8; off >= 1; off >>= 1) {
            float ov = __shfl_xor(bv, off, 32);
            int   oi = __shfl_xor(bi, off, 32);
            if (ov > bv || (ov == bv && oi < bi)) { bv = ov; bi = oi; }
        }
        if (m == 0) preds[row] = bi;
    }
}

// ---------------------------------------------------------------------------
// Kernel 2: masked segment-sum update.  Block owns DPB dims; LDS accumulator
// lacc[26][DPB] (26.6 KB << 320 KB/WGP).  Row range split ROWCHUNKS ways;
// partial sums merged with global f32 atomics into ws accumulator.
// ---------------------------------------------------------------------------
__global__ void __launch_bounds__(DPB)
accumulate_updates_kernel(const float* __restrict__ enc,
                          const int*   __restrict__ targets,
                          const int*   __restrict__ preds,
                          float* __restrict__ acc) {
    __shared__ float lacc[NCLASSES * DPB];
    const int tl  = threadIdx.x;
    const int dim = blockIdx.x * DPB + tl;
    const bool dv = (dim < DHV);

    for (int i = tl; i < NCLASSES * DPB; i += blockDim.x) lacc[i] = 0.0f;
    __syncthreads();

    const int rowsPer = NROWS / ROWCHUNKS;
    const int rbeg    = blockIdx.y * rowsPer;
    for (int r = rbeg; r < rbeg + rowsPer; ++r) {
        const int p = preds[r];
        const int t = targets[r];
        if (p != t && dv) {
            const float e = enc[(size_t)r * DHV + dim];
            atomicAdd(&lacc[t * DPB + tl],  e);   // ds_add_f32 (no contention)
            atomicAdd(&lacc[p * DPB + tl], -e);
        }
    }
    __syncthreads();

    if (dv) {
        for (int c = 0; c < NCLASSES; ++c)
            atomicAdd(&acc[(size_t)c * DHV + dim], lacc[c * DPB + tl]);
    }
}

// ---------------------------------------------------------------------------
// Kernel 3: new_class_hvs = class_hvs + acc
// ---------------------------------------------------------------------------
__global__ void finalize_kernel(const float* __restrict__ chv,
                                const float* __restrict__ acc,
                                float* __restrict__ out, int n) {
    int i = blockIdx.x * blockDim.x + threadIdx.x;
    if (i < n) out[i] = chv[i] + acc[i];
}

// ---------------------------------------------------------------------------
extern "C" void kernel_launch(void* const* d_in, const int* in_sizes, int n_in,
                              void* d_out, int out_size, void* d_ws, size_t ws_size,
                              hipStream_t stream) {
    const float* enc     = (const float*)d_in[0];   // [N, D] f32
    const int*   targets = (const int*)  d_in[1];   // [N] i32
    const float* chv     = (const float*)d_in[2];   // [26, D] f32

    float* scores = (float*)d_out;                                   // [N, 26]
    float* newchv = (float*)d_out + (size_t)NROWS * NCLASSES;        // [26, D]

    int*   preds = (int*)d_ws;                                       // [N] i32
    float* acc   = (float*)((char*)d_ws + (size_t)NROWS * sizeof(int)); // [26, D]

    (void)in_sizes; (void)n_in; (void)out_size; (void)ws_size;

    const int accN = NCLASSES * DHV;
    zero_acc_kernel<<<(accN + 255) / 256, 256, 0, stream>>>(acc, accN);

    // 512 row-tiles of 16 rows, 8 waves (256 threads) per block -> 64 blocks
    gemm_scores_preds_kernel<<<(NROWS / 16) / 8, 256, 0, stream>>>(
        enc, chv, scores, preds);

    dim3 ugrid((DHV + DPB - 1) / DPB, ROWCHUNKS);
    accumulate_updates_kernel<<<ugrid, DPB, 0, stream>>>(enc, targets, preds, acc);

    finalize_kernel<<<(accN + 255) / 256, 256, 0, stream>>>(chv, acc, newchv, accN);
}